// QueryAndGroup_55327768707540
// MI455X (gfx1250) — compile-verified
//
#include <hip/hip_runtime.h>
#include <hip/hip_bf16.h>

// Problem constants (from reference setup_inputs)
#define BB      4
#define NN      16384
#define NPT     2048
#define CC      64
#define KNN     32
#define RADIUS_ 0.5f
#define CHUNK   128      // points staged in LDS per iteration (8 waves x 16)
#define QT      16       // queries per workgroup tile (one WMMA M dimension)

typedef float v2f __attribute__((ext_vector_type(2)));
typedef float v8f __attribute__((ext_vector_type(8)));

// ---------------------------------------------------------------------------
// Kernel 1: fused WMMA distance + per-query top-32 selection + radius mask.
// One workgroup = 16 queries, 8 waves. Wave w computes the 16x16 cross tile
// for its 16-point slice with V_WMMA_F32_16X16X4_F32 (A = queries 16x4,
// zero-padded K; B = points 4x16), then waves select top-32 for 2 queries
// each with ballot-filtered insertions against a wave-max threshold.
// Score tracked = |p|^2 - 2 q.p  (|q|^2 is constant per query -> order
// invariant; added back only for the radius test).
// Point chunks are double-buffered into LDS with the CDNA5 async path
// (GLOBAL_LOAD_ASYNC_TO_LDS_B128, ASYNCcnt): chunk i+1's HBM/L2 fetch
// overlaps chunk i's WMMA + selection, with no VGPR staging.
// ---------------------------------------------------------------------------
__global__ __launch_bounds__(256) void knn_select_kernel(
    const float* __restrict__ xyz,      // (B, N, 3)
    const float* __restrict__ new_xyz,  // (B, NPT, 3)
    int* __restrict__ out_idx)          // (B, NPT, KNN)
{
    __shared__ __align__(16) float s_raw[2][CHUNK * 3];  // double-buffered chunk
    __shared__ float s_scores[QT * 132];                 // 16 x 128 (+pad) scores
    __shared__ float s_qx[QT], s_qy[QT], s_qz[QT], s_q2[QT];

    const int t    = threadIdx.x;
    const int lane = t & 31;
    const int w    = t >> 5;                  // wave id 0..7
    const int tile = blockIdx.x & (NPT / QT - 1);
    const int b    = blockIdx.x / (NPT / QT);

    const float* xyzb = xyz + (size_t)b * NN * 3;

    // Issue one async B128 global->LDS copy per lane (96 lanes cover
    // 128 pts * 12 B = 1536 B). VDST = per-lane LDS byte address,
    // VADDR = per-lane byte offset, SADDR = 64-bit base (GVS mode).
#define ASYNC_STAGE(CBN, BUFP)                                               \
    do {                                                                     \
        if (t < 96) {                                                        \
            unsigned _lds = (unsigned)(size_t)((BUFP) + t * 4);              \
            unsigned _off = (unsigned)(((CBN) * 3 + t * 4) * 4);             \
            asm volatile("global_load_async_to_lds_b128 %0, %1, %2 offset:0" \
                         :: "v"(_lds), "v"(_off), "s"(xyzb) : "memory");     \
        }                                                                    \
    } while (0)

    if (t < QT) {
        const float* q = new_xyz + ((size_t)b * NPT + tile * QT + t) * 3;
        const float x = q[0], y = q[1], z = q[2];
        s_qx[t] = x; s_qy[t] = y; s_qz[t] = z;
        s_q2[t] = x * x + y * y + z * z;
    }

    // Prologue: start fetching chunk 0.
    ASYNC_STAGE(0, s_raw[0]);
    __syncthreads();

    // Loop-invariant A matrix (16 queries x 4, K3 = 0 pad).
    // Layout (ISA 7.12.2, 32-bit A 16x4): lanes 0-15: VGPR0=K0, VGPR1=K1;
    // lanes 16-31: VGPR0=K2, VGPR1=K3.
    const int m16 = lane & 15;
    v2f A;
    A.x = (lane < 16) ? s_qx[m16] : s_qz[m16];
    A.y = (lane < 16) ? s_qy[m16] : 0.0f;

    // Per-lane candidate store: 2 queries per wave (q = 2w, 2w+1).
    float bd[2] = { __builtin_inff(), __builtin_inff() };
    int   bi[2] = { 0, 0 };
    float th[2] = { __builtin_inff(), __builtin_inff() };

    for (int cb = 0, it = 0; cb < NN; cb += CHUNK, ++it) {
        const float* curb = s_raw[it & 1];

        if (cb + CHUNK < NN) {
            // Kick off next chunk into the other buffer, then wait until
            // only it remains outstanding (async loads complete in order,
            // so <=1 means the current chunk has landed).
            ASYNC_STAGE(cb + CHUNK, s_raw[(it + 1) & 1]);
            asm volatile("s_wait_asynccnt 0x1" ::: "memory");
        } else {
            asm volatile("s_wait_asynccnt 0x0" ::: "memory");
        }
        __syncthreads();   // current buffer visible to all waves

        // This wave's 16-point slice -> B matrix (4x16, K-major mirrored).
        const int   ptl = w * 16 + m16;
        const float px  = curb[3 * ptl + 0];
        const float py  = curb[3 * ptl + 1];
        const float pz  = curb[3 * ptl + 2];
        v2f Bm;
        Bm.x = (lane < 16) ? px : pz;
        Bm.y = (lane < 16) ? py : 0.0f;
        const float p2 = px * px + py * py + pz * pz;

        v8f acc = {0.f, 0.f, 0.f, 0.f, 0.f, 0.f, 0.f, 0.f};
        // (neg_a, A, neg_b, B, c_mod, C, reuse_a, reuse_b)
        acc = __builtin_amdgcn_wmma_f32_16x16x4_f32(
                  false, A, false, Bm, (short)0, acc, false, false);

        // D layout: VGPR j -> M = j + 8*(lane>=16), N = lane&15.
        const int mbase = (lane >> 4) << 3;
        #pragma unroll
        for (int j = 0; j < 8; ++j)
            s_scores[(mbase + j) * 132 + ptl] = fmaf(-2.0f, acc[j], p2);
        __syncthreads();

        // Selection: each wave maintains top-32 for queries 2w, 2w+1.
        for (int qi = 0; qi < 2; ++qi) {
            const int q   = 2 * w + qi;
            float thq = th[qi];
            float bdq = bd[qi];
            int   biq = bi[qi];
            for (int g = 0; g < 4; ++g) {
                const int   local = g * 32 + lane;
                const float v     = s_scores[q * 132 + local];
                const int   pidx  = cb + local;
                unsigned long long mball = __ballot(v < thq) & 0xffffffffull;
                while (mball) {
                    const int src = __builtin_ctzll(mball);
                    mball &= mball - 1;
                    const float cv = __shfl(v, src);
                    const int   ci = __shfl(pidx, src);
                    if (!(cv < thq)) continue;   // threshold tightened
                    // wave argmax over held candidates
                    float mv = bdq; int ml = lane;
                    #pragma unroll
                    for (int off = 16; off > 0; off >>= 1) {
                        const float ov = __shfl_xor(mv, off);
                        const int   ol = __shfl_xor(ml, off);
                        if (ov > mv || (ov == mv && ol > ml)) { mv = ov; ml = ol; }
                    }
                    if (lane == ml) { bdq = cv; biq = ci; }
                    // refresh threshold = wave max of candidates
                    float t2 = bdq;
                    #pragma unroll
                    for (int off = 16; off > 0; off >>= 1)
                        t2 = fmaxf(t2, __shfl_xor(t2, off));
                    thq = t2;
                }
            }
            th[qi] = thq; bd[qi] = bdq; bi[qi] = biq;
        }
        __syncthreads();
    }
#undef ASYNC_STAGE

    // Sort ascending (score, idx) across the 32 lanes, apply radius mask.
    for (int qi = 0; qi < 2; ++qi) {
        const int q = 2 * w + qi;
        float d  = bd[qi];
        int   ix = bi[qi];
        #pragma unroll
        for (int k = 2; k <= 32; k <<= 1) {
            #pragma unroll
            for (int j = k >> 1; j > 0; j >>= 1) {
                const float od = __shfl_xor(d, j);
                const int   oi = __shfl_xor(ix, j);
                const bool up       = ((lane & k) == 0);
                const bool mine_min = (d < od) || (d == od && ix < oi);
                const bool want_min = (((lane & j) == 0) == up);
                if (want_min != mine_min) { d = od; ix = oi; }
            }
        }
        const int   nearest = __shfl(ix, 0);
        const float d2full  = fmaxf(d + s_q2[q], 0.0f);
        const float dist    = sqrtf(d2full);
        const int   outv    = (dist > RADIUS_) ? nearest : ix;
        out_idx[((size_t)b * NPT + tile * QT + q) * KNN + lane] = outv;
    }
}

// ---------------------------------------------------------------------------
// Kernel 2: gather + recenter + write both outputs. One 32-lane group per
// (b, p); lane = sample. Writes coalesced along the sample axis; feature
// gathers hit L2 (features row = 64KB, idx clusters spatially).
// out = [ new_features (B,67,NPT,KNN) | grouped_xyz (B,3,NPT,KNN) ]
// ---------------------------------------------------------------------------
__global__ __launch_bounds__(256) void gather_kernel(
    const float* __restrict__ xyz,
    const float* __restrict__ new_xyz,
    const float* __restrict__ features,   // (B, C, N)
    const int*   __restrict__ idx,        // (B, NPT, KNN)
    float* __restrict__ out)
{
    const int lane   = threadIdx.x;               // sample 0..31
    const int grp    = threadIdx.y;               // 0..7
    const int linear = blockIdx.x * 8 + grp;      // b*NPT + p
    const int b      = linear / NPT;
    const int p      = linear - b * NPT;

    const int id = idx[(size_t)linear * KNN + lane];

    const float* xb = xyz + ((size_t)b * NN + id) * 3;
    const float* qb = new_xyz + (size_t)linear * 3;
    const float gx = xb[0] - qb[0];
    const float gy = xb[1] - qb[1];
    const float gz = xb[2] - qb[2];

    const size_t cs    = (size_t)NPT * KNN;
    const size_t base1 = ((size_t)b * (CC + 3) * NPT + p) * KNN + lane;
    out[base1 + 0 * cs] = gx;
    out[base1 + 1 * cs] = gy;
    out[base1 + 2 * cs] = gz;

    const size_t g2base = (size_t)BB * (CC + 3) * NPT * KNN;
    const size_t base2  = g2base + ((size_t)b * 3 * NPT + p) * KNN + lane;
    out[base2 + 0 * cs] = gx;
    out[base2 + 1 * cs] = gy;
    out[base2 + 2 * cs] = gz;

    const float* fb = features + (size_t)b * CC * NN + id;
    #pragma unroll 4
    for (int c = 0; c < CC; ++c)
        out[base1 + (size_t)(3 + c) * cs] = fb[(size_t)c * NN];
}

extern "C" void kernel_launch(void* const* d_in, const int* in_sizes, int n_in,
                              void* d_out, int out_size, void* d_ws, size_t ws_size,
                              hipStream_t stream) {
    (void)in_sizes; (void)n_in; (void)out_size; (void)ws_size;
    const float* xyz      = (const float*)d_in[0];
    const float* new_xyz  = (const float*)d_in[1];
    const float* features = (const float*)d_in[2];
    float* out    = (float*)d_out;
    int*   ws_idx = (int*)d_ws;   // needs B*NPT*KNN*4 = 1 MB scratch

    knn_select_kernel<<<dim3(BB * (NPT / QT)), 256, 0, stream>>>(
        xyz, new_xyz, ws_idx);

    gather_kernel<<<dim3((BB * NPT) / 8), dim3(32, 8), 0, stream>>>(
        xyz, new_xyz, features, ws_idx, out);
}